// Attention_59691455480358
// MI455X (gfx1250) — compile-verified
//
#include <hip/hip_runtime.h>
#include <math.h>

typedef __attribute__((ext_vector_type(2))) float v2f;
typedef __attribute__((ext_vector_type(4))) float v4f;
typedef __attribute__((ext_vector_type(8))) float v8f;

#define NROW 2049
#define CDIM 1024
#define C3   3072
#define NB   16
#define NOUT 2048

// Slot mapping note: V_WMMA_F32_16X16X4_F32 sums A[m][slot]*B[slot][n] over 4
// K-slots: (laneGroup g, vgpr j). We fill slot (g,j) with physical k = k0+4g+j
// for WMMA0 and k = k0+4g+j+2 for WMMA1, identically in A and B, so each lane
// feeds both WMMAs from ONE b128 load of x[m, k0+4g .. k0+4g+3].

// ---------------------------------------------------------------------------
// K1: Q (16x1024) = X0 (16x1024) @ W_q (1024x1024)
// ---------------------------------------------------------------------------
__global__ __launch_bounds__(32) void qcls_kernel(const float* __restrict__ x,
                                                  const float* __restrict__ w,
                                                  float* __restrict__ Q) {
  const int lane = threadIdx.x;
  const int col  = lane & 15;   // N index (and A row index m)
  const int half = lane >> 4;   // lane group g
  const int n0   = blockIdx.x * 16;

  v8f acc = {};
  for (int k = 0; k < CDIM; k += 8) {
    // A: x[batch=col, 0, k + 4*half .. +3]  (one b128 load)
    v4f a4 = *(const v4f*)(x + (size_t)col * (NROW * CDIM) + k + 4 * half);
    // B slot (half, j) for WMMA0 = W_q[k+4*half+j][n0+col]; WMMA1 = +2 rows
    const float* bp = w + (size_t)(k + 4 * half) * C3 + (n0 + col);
    v2f a0; a0.x = a4.x; a0.y = a4.y;
    v2f a1; a1.x = a4.z; a1.y = a4.w;
    v2f b0; b0.x = bp[0];          b0.y = bp[C3];
    v2f b1; b1.x = bp[2 * (size_t)C3]; b1.y = bp[3 * (size_t)C3];
    acc = __builtin_amdgcn_wmma_f32_16x16x4_f32(false, a0, false, b0,
                                                (short)0, acc, false, false);
    acc = __builtin_amdgcn_wmma_f32_16x16x4_f32(false, a1, false, b1,
                                                (short)0, acc, false, false);
  }
  #pragma unroll
  for (int r = 0; r < 8; ++r)
    Q[(r + 8 * half) * CDIM + n0 + col] = acc[r];
}

// ---------------------------------------------------------------------------
// K2: V (16x1024) = Q (16x1024) @ W_k^T   (W_k = w[:, 1024:2048])
// v[b,d] = sum_c Q[b,c] * w[d*3072 + 1024 + c] — both operands b128-loadable.
// ---------------------------------------------------------------------------
__global__ __launch_bounds__(32) void vvec_kernel(const float* __restrict__ w,
                                                  const float* __restrict__ Q,
                                                  float* __restrict__ V) {
  const int lane = threadIdx.x;
  const int col  = lane & 15;
  const int half = lane >> 4;
  const int d0   = blockIdx.x * 16;

  v8f acc = {};
  for (int k = 0; k < CDIM; k += 8) {
    v4f a4 = *(const v4f*)(Q + col * CDIM + k + 4 * half);
    v4f b4 = *(const v4f*)(w + (size_t)(d0 + col) * C3 + CDIM + k + 4 * half);
    v2f a0; a0.x = a4.x; a0.y = a4.y;
    v2f a1; a1.x = a4.z; a1.y = a4.w;
    v2f b0; b0.x = b4.x; b0.y = b4.y;
    v2f b1; b1.x = b4.z; b1.y = b4.w;
    acc = __builtin_amdgcn_wmma_f32_16x16x4_f32(false, a0, false, b0,
                                                (short)0, acc, false, false);
    acc = __builtin_amdgcn_wmma_f32_16x16x4_f32(false, a1, false, b1,
                                                (short)0, acc, false, false);
  }
  #pragma unroll
  for (int r = 0; r < 8; ++r)
    V[(r + 8 * half) * CDIM + d0 + col] = acc[r];
}

// ---------------------------------------------------------------------------
// K3: logits[b, j] = x[b, j+1, :] . V[b, :] * C^-0.5  — the 134 MB pass.
// One b128 load per lane feeds two WMMAs (8 K-steps). v_b occupies B column 0.
// ---------------------------------------------------------------------------
__global__ __launch_bounds__(128) void logits_kernel(const float* __restrict__ x,
                                                     const float* __restrict__ V,
                                                     float* __restrict__ out) {
  const int lane = threadIdx.x & 31;
  const int wave = threadIdx.x >> 5;
  const int col  = lane & 15;
  const int half = lane >> 4;
  const int b    = blockIdx.x >> 5;               // batch
  const int tile = (blockIdx.x & 31) * 4 + wave;  // 0..127
  const int n0   = tile * 16;                     // x rows n0+1 .. n0+16

  const float* xb = x + (size_t)b * (NROW * CDIM) + (size_t)(n0 + 1) * CDIM;
  const float* vb = V + b * CDIM;

  v8f acc = {};
  for (int k = 0; k < CDIM; k += 8) {
    v4f a4 = *(const v4f*)(xb + (size_t)col * CDIM + k + 4 * half);
    v4f v4 = *(const v4f*)(vb + k + 4 * half);   // uniform per lane group, L0 hit
    v2f a0; a0.x = a4.x; a0.y = a4.y;
    v2f a1; a1.x = a4.z; a1.y = a4.w;
    v2f b0; b0.x = (col == 0) ? v4.x : 0.0f; b0.y = (col == 0) ? v4.y : 0.0f;
    v2f b1; b1.x = (col == 0) ? v4.z : 0.0f; b1.y = (col == 0) ? v4.w : 0.0f;
    acc = __builtin_amdgcn_wmma_f32_16x16x4_f32(false, a0, false, b0,
                                                (short)0, acc, false, false);
    acc = __builtin_amdgcn_wmma_f32_16x16x4_f32(false, a1, false, b1,
                                                (short)0, acc, false, false);
  }
  // D column 0 lives in lane 0 (rows 0..7) and lane 16 (rows 8..15)
  if (col == 0) {
    const float scale = 0.03125f;  // 1024^-0.5
    #pragma unroll
    for (int r = 0; r < 8; ++r)
      out[b * NOUT + n0 + r + 8 * half] = acc[r] * scale;
  }
}

// ---------------------------------------------------------------------------
// K4: in-place softmax over each row of 2048 (16 rows). Two-pass LDS reduce.
// ---------------------------------------------------------------------------
__global__ __launch_bounds__(256) void softmax_kernel(float* __restrict__ out) {
  __shared__ float red[256];
  const int b = blockIdx.x;
  const int t = threadIdx.x;
  float* row = out + b * NOUT;

  float vals[8];
  float m = -INFINITY;
  #pragma unroll
  for (int i = 0; i < 8; ++i) {
    vals[i] = row[t + i * 256];
    m = fmaxf(m, vals[i]);
  }
  red[t] = m;
  __syncthreads();
  for (int s = 128; s > 0; s >>= 1) {
    if (t < s) red[t] = fmaxf(red[t], red[t + s]);
    __syncthreads();
  }
  m = red[0];
  __syncthreads();

  float sum = 0.0f;
  #pragma unroll
  for (int i = 0; i < 8; ++i) {
    vals[i] = __expf(vals[i] - m);
    sum += vals[i];
  }
  red[t] = sum;
  __syncthreads();
  for (int s = 128; s > 0; s >>= 1) {
    if (t < s) red[t] += red[t + s];
    __syncthreads();
  }
  const float inv = 1.0f / red[0];
  #pragma unroll
  for (int i = 0; i < 8; ++i)
    row[t + i * 256] = vals[i] * inv;
}

// ---------------------------------------------------------------------------
extern "C" void kernel_launch(void* const* d_in, const int* in_sizes, int n_in,
                              void* d_out, int out_size, void* d_ws, size_t ws_size,
                              hipStream_t stream) {
  const float* x = (const float*)d_in[0];   // (16, 2049, 1024) f32
  const float* w = (const float*)d_in[1];   // (1024, 3072) f32
  float* out = (float*)d_out;               // (16, 2048) f32

  float* Q = (float*)d_ws;                  // 16*1024 f32 = 64 KB
  float* V = Q + NB * CDIM;                 // 16*1024 f32 = 64 KB

  qcls_kernel<<<64, 32, 0, stream>>>(x, w, Q);
  vvec_kernel<<<64, 32, 0, stream>>>(w, Q, V);
  logits_kernel<<<NB * 32, 128, 0, stream>>>(x, V, out);
  softmax_kernel<<<NB, 256, 0, stream>>>(out);
}